// ContextualAttention_86165633892916
// MI455X (gfx1250) — compile-verified
//
#include <hip/hip_runtime.h>
#include <hip/hip_bf16.h>

typedef __attribute__((ext_vector_type(2))) float v2f;
typedef __attribute__((ext_vector_type(8))) float v8f;
typedef __attribute__((ext_vector_type(4))) int   v4i;

#if defined(__has_builtin)
#if __has_builtin(__builtin_amdgcn_global_load_async_to_lds_b128) && \
    __has_builtin(__builtin_amdgcn_s_wait_asynccnt)
#define ASYNC_A 1
#endif
#endif
#ifndef ASYNC_A
#define ASYNC_A 0
#endif

#if ASYNC_A
typedef __attribute__((address_space(1))) v4i gv4i;
typedef __attribute__((address_space(3))) v4i lv4i;
__device__ __forceinline__ gv4i* to_global(const void* p) {
  return (gv4i*)(unsigned long long)p;           // generic->AS1 via integer
}
__device__ __forceinline__ lv4i* to_lds(void* p) {
  // low 32 bits of a generic LDS address are the LDS byte offset (ISA 10.2)
  return (lv4i*)(unsigned int)(unsigned long long)p;
}
#endif

struct Epi {
  const float* bias;     // MODE1/2: per-col bias
  const float* bvec;     // MODE2: per-batch row vector (gctx)
  int bvecRowDiv;        // rows per batch (S)
  int bvecStride;        // D
  const float* rowbias;  // MODE3: per-row additive bias
  float alpha;           // MODE3: scale
};

struct ZMap {  // blockIdx.z -> pointer offsets: z = zh*zsplit + zl
  int zsplit;
  long long aHi, aLo, bHi, bLo, cHi, cLo, rHi, rLo;
};

// C[M,N] = A[M,K] @ B[K,N] (+epilogue).
// 256 threads = 8 waves; wave tile 32x64 (2 A-frags x 4 B-frags, 8 accums).
// Block tile (32*WM) x (64*WN). BTRANS: logical B[k][n] = Bm[n*ldb + k].
// Double-buffered LDS, one barrier per K-step. A tile staged via async
// global->LDS (ASYNCcnt) when available, else register staging. B staged via
// registers into a k-pair-interleaved layout so each fragment is one b64 load.
// MODE 0: store; 1: +bias[col]; 2: sigmoid(acc+bias[col]+bvec[b][col]); 3: acc*alpha+rowbias[row]
template<int WM, int WN, bool BTRANS, int MODE>
__global__ __launch_bounds__(256) void gemm_wmma_f32(
    const float* __restrict__ A, int lda,
    const float* __restrict__ Bm, int ldb,
    float* __restrict__ C, int ldc,
    int K, Epi e, ZMap zm)
{
  constexpr int BM  = 32 * WM;
  constexpr int BN  = 64 * WN;
  constexpr int BNP = BN + 16;          // pad: kp vs kp+1 rows 32 banks apart
  constexpr int AP  = 20;               // A row pad: 80B stride (16B aligned), conflict-free
  constexpr int AIT = (BM * 16) / 1024; // float4 A transfers per thread
  constexpr int BIT = BTRANS ? (BN * 4) / 256 : (16 * BN) / 1024;

  const float* rowbias = e.rowbias;
  if (zm.zsplit > 0) {
    const long long zh = blockIdx.z / zm.zsplit;
    const long long zl = blockIdx.z % zm.zsplit;
    A  += zh * zm.aHi + zl * zm.aLo;
    Bm += zh * zm.bHi + zl * zm.bLo;
    C  += zh * zm.cHi + zl * zm.cLo;
    if (MODE == 3) rowbias += zh * zm.rHi + zl * zm.rLo;
  }
  const int m0 = blockIdx.y * BM;
  const int n0 = blockIdx.x * BN;

  __shared__ float As[2][BM][AP];        // [buf][row][k]
  __shared__ float Bs[2][8][BNP][2];     // [buf][kpair][n][k&1]

  const int tid  = threadIdx.x;
  const int lane = tid & 31;
  const int wave = tid >> 5;
  const int wm = wave % WM;
  const int wn = wave / WM;
  const int r  = lane & 15;
  const int kh = lane >> 4;              // K-half select per ISA frag layout

  v8f acc[2][4];
#pragma unroll
  for (int mi = 0; mi < 2; ++mi)
#pragma unroll
    for (int j = 0; j < 4; ++j)
#pragma unroll
      for (int i = 0; i < 8; ++i) acc[mi][j][i] = 0.f;

#if !ASYNC_A
  float4 aReg[AIT];
#endif
  float4 bReg[BIT];

  // ---- prologue: stage tile k0=0 ----
#if ASYNC_A
#pragma unroll
  for (int i = 0; i < AIT; ++i) {
    const int idx = tid * 4 + i * 1024;
    const int row = idx >> 4, col = idx & 15;
    __builtin_amdgcn_global_load_async_to_lds_b128(
        to_global(A + (long long)(m0 + row) * lda + col),
        to_lds(&As[0][row][col]), 0, 0);
  }
#else
#pragma unroll
  for (int i = 0; i < AIT; ++i) {
    const int idx = tid * 4 + i * 1024;
    const int row = idx >> 4, col = idx & 15;
    aReg[i] = *(const float4*)(A + (long long)(m0 + row) * lda + col);
  }
#endif
  if constexpr (!BTRANS) {
#pragma unroll
    for (int i = 0; i < BIT; ++i) {
      const int idx = tid * 4 + i * 1024;
      const int row = idx / BN, col = idx % BN;
      bReg[i] = *(const float4*)(Bm + (long long)row * ldb + (n0 + col));
    }
  } else {
#pragma unroll
    for (int i = 0; i < BIT; ++i) {
      const int t = tid + i * 256;
      const int n = t >> 2, kq = (t & 3) * 4;
      bReg[i] = *(const float4*)(Bm + (long long)(n0 + n) * ldb + kq);
    }
  }

  int buf = 0;
  for (int k0 = 0; k0 < K; k0 += 16) {
    // ---- commit staged data into LDS buffer buf ----
#if !ASYNC_A
#pragma unroll
    for (int i = 0; i < AIT; ++i) {
      const int idx = tid * 4 + i * 1024;
      const int row = idx >> 4, col = idx & 15;
      As[buf][row][col + 0] = aReg[i].x; As[buf][row][col + 1] = aReg[i].y;
      As[buf][row][col + 2] = aReg[i].z; As[buf][row][col + 3] = aReg[i].w;
    }
#endif
    if constexpr (!BTRANS) {
#pragma unroll
      for (int i = 0; i < BIT; ++i) {
        const int idx = tid * 4 + i * 1024;
        const int row = idx / BN, col = idx % BN;
        const int kp = row >> 1, kl = row & 1;
        Bs[buf][kp][col + 0][kl] = bReg[i].x; Bs[buf][kp][col + 1][kl] = bReg[i].y;
        Bs[buf][kp][col + 2][kl] = bReg[i].z; Bs[buf][kp][col + 3][kl] = bReg[i].w;
      }
    } else {
#pragma unroll
      for (int i = 0; i < BIT; ++i) {
        const int t = tid + i * 256;
        const int n = t >> 2, kq = (t & 3) * 4;
        Bs[buf][(kq >> 1) + 0][n][0] = bReg[i].x; Bs[buf][(kq >> 1) + 0][n][1] = bReg[i].y;
        Bs[buf][(kq >> 1) + 1][n][0] = bReg[i].z; Bs[buf][(kq >> 1) + 1][n][1] = bReg[i].w;
      }
    }
#if ASYNC_A
    __builtin_amdgcn_s_wait_asynccnt(0);   // A tile in LDS buffer buf has landed
#endif
    __syncthreads();

    // ---- issue next tile's loads (overlap with WMMA below) ----
    const int kn = k0 + 16;
    if (kn < K) {
#if ASYNC_A
#pragma unroll
      for (int i = 0; i < AIT; ++i) {
        const int idx = tid * 4 + i * 1024;
        const int row = idx >> 4, col = idx & 15;
        __builtin_amdgcn_global_load_async_to_lds_b128(
            to_global(A + (long long)(m0 + row) * lda + (kn + col)),
            to_lds(&As[buf ^ 1][row][col]), 0, 0);
      }
#else
#pragma unroll
      for (int i = 0; i < AIT; ++i) {
        const int idx = tid * 4 + i * 1024;
        const int row = idx >> 4, col = idx & 15;
        aReg[i] = *(const float4*)(A + (long long)(m0 + row) * lda + (kn + col));
      }
#endif
      if constexpr (!BTRANS) {
#pragma unroll
        for (int i = 0; i < BIT; ++i) {
          const int idx = tid * 4 + i * 1024;
          const int row = idx / BN, col = idx % BN;
          const float* src = Bm + (long long)(kn + row) * ldb + (n0 + col);
          bReg[i] = *(const float4*)src;
          if (kn + 16 < K) __builtin_prefetch(src + 16 * ldb, 0, 1);
        }
      } else {
#pragma unroll
        for (int i = 0; i < BIT; ++i) {
          const int t = tid + i * 256;
          const int n = t >> 2, kq = (t & 3) * 4;
          const float* src = Bm + (long long)(n0 + n) * ldb + (kn + kq);
          bReg[i] = *(const float4*)src;
          if (kn + 16 < K) __builtin_prefetch(src + 16, 0, 1);
        }
      }
    }

    // ---- load ALL fragments first, then fire 32 WMMAs back-to-back ----
    v2f afrag[4][2];
    v2f bfrag[4][4];
#pragma unroll
    for (int kk = 0; kk < 4; ++kk) {
      const int kb = kk * 4 + kh * 2;
      const int kp = kk * 2 + kh;
      afrag[kk][0] = *(const v2f*)&As[buf][wm * 32 + r][kb];
      afrag[kk][1] = *(const v2f*)&As[buf][wm * 32 + 16 + r][kb];
#pragma unroll
      for (int j = 0; j < 4; ++j)
        bfrag[kk][j] = *(const v2f*)&Bs[buf][kp][wn * 64 + j * 16 + r][0];
    }
#pragma unroll
    for (int kk = 0; kk < 4; ++kk) {
#pragma unroll
      for (int j = 0; j < 4; ++j) {
        acc[0][j] = __builtin_amdgcn_wmma_f32_16x16x4_f32(
            false, afrag[kk][0], false, bfrag[kk][j], (short)0, acc[0][j], false, false);
        acc[1][j] = __builtin_amdgcn_wmma_f32_16x16x4_f32(
            false, afrag[kk][1], false, bfrag[kk][j], (short)0, acc[1][j], false, false);
      }
    }
    buf ^= 1;
  }

  // ---- epilogue + store (C frag: lane col=r, VGPR i -> row i + 8*kh) ----
#pragma unroll
  for (int mi = 0; mi < 2; ++mi) {
#pragma unroll
    for (int j = 0; j < 4; ++j) {
      const int col = n0 + wn * 64 + j * 16 + r;
#pragma unroll
      for (int i = 0; i < 8; ++i) {
        const int row = m0 + wm * 32 + mi * 16 + kh * 8 + i;
        float v = acc[mi][j][i];
        if constexpr (MODE == 1) v += e.bias[col];
        if constexpr (MODE == 2) {
          v += e.bias[col] + e.bvec[(row / e.bvecRowDiv) * e.bvecStride + col];
          v = 1.f / (1.f + __expf(-v));
        }
        if constexpr (MODE == 3) v = v * e.alpha + rowbias[row];
        C[(long long)row * ldc + col] = v;
      }
    }
  }
}

// ctx_proj[b][n] = context[b] . Wc[:,n] + bc[n]   (4096 threads)
__global__ __launch_bounds__(256) void ctxproj_kernel(
    const float* __restrict__ ctx, const float* __restrict__ Wc,
    const float* __restrict__ bc, float* __restrict__ cproj)
{
  const int idx = blockIdx.x * 256 + threadIdx.x;
  const int b = idx >> 10, col = idx & 1023;
  float s = bc[col];
  for (int k = 0; k < 256; ++k) s = fmaf(ctx[b * 256 + k], Wc[k * 1024 + col], s);
  cproj[idx] = s;
}

// gctx[b][n] = ctx_proj[b] . Wg[D:2D, n]   (per-batch half of the gate GEMM)
__global__ __launch_bounds__(256) void gctx_kernel(
    const float* __restrict__ cproj, const float* __restrict__ Wg,
    float* __restrict__ gctx)
{
  const int idx = blockIdx.x * 256 + threadIdx.x;
  const int b = idx >> 10, col = idx & 1023;
  float s = 0.f;
  for (int k = 0; k < 1024; ++k)
    s = fmaf(cproj[b * 1024 + k], Wg[(1024 + k) * 1024 + col], s);
  gctx[idx] = s;
}

// g[b][h][s] = mean_d gate[b][s][h*64+d]   (65536 threads)
__global__ __launch_bounds__(256) void greduce_kernel(
    const float* __restrict__ gate, float* __restrict__ g)
{
  const int idx = blockIdx.x * 256 + threadIdx.x;
  const int b = idx >> 14, h = (idx >> 10) & 15, s = idx & 1023;
  const float* p = gate + ((long long)(b * 1024 + s)) * 1024 + h * 64;
  float acc = 0.f;
  for (int d = 0; d < 64; ++d) acc += p[d];
  g[idx] = acc * (1.f / 64.f);
}

// rowbias[b][h][i] = opt_bias[h] + (Q[b,i,h*64:] . ctx_head[b,h]) / 8 * g[b,h,i]
__global__ __launch_bounds__(256) void rowbias_kernel(
    const float* __restrict__ Q, const float* __restrict__ cproj,
    const float* __restrict__ g, const float* __restrict__ ob,
    float* __restrict__ rowb)
{
  const int idx = blockIdx.x * 256 + threadIdx.x;
  const int b = idx >> 14, h = (idx >> 10) & 15, i = idx & 1023;
  const float* q = Q + ((long long)(b * 1024 + i)) * 1024 + h * 64;
  const float* c = cproj + b * 1024 + h * 64;
  float s = 0.f;
  for (int d = 0; d < 64; ++d) s = fmaf(q[d], c[d], s);
  rowb[idx] = ob[h] + s * 0.125f * g[idx];
}

// in-place softmax over rows of 1024 (one 256-thread block per row)
__global__ __launch_bounds__(256) void softmax_kernel(float* __restrict__ attn)
{
  const long long row = blockIdx.x;
  float* p = attn + row * 1024;
  const int t = threadIdx.x;
  float4 v = ((float4*)p)[t];
  __shared__ float red[256];
  float m = fmaxf(fmaxf(v.x, v.y), fmaxf(v.z, v.w));
  red[t] = m; __syncthreads();
  for (int s = 128; s > 0; s >>= 1) {
    if (t < s) red[t] = fmaxf(red[t], red[t + s]);
    __syncthreads();
  }
  m = red[0]; __syncthreads();
  v.x = __expf(v.x - m); v.y = __expf(v.y - m);
  v.z = __expf(v.z - m); v.w = __expf(v.w - m);
  red[t] = v.x + v.y + v.z + v.w; __syncthreads();
  for (int s = 128; s > 0; s >>= 1) {
    if (t < s) red[t] += red[t + s];
    __syncthreads();
  }
  const float inv = 1.f / red[0];
  v.x *= inv; v.y *= inv; v.z *= inv; v.w *= inv;
  ((float4*)p)[t] = v;
}

extern "C" void kernel_launch(void* const* d_in, const int* in_sizes, int n_in,
                              void* d_out, int out_size, void* d_ws, size_t ws_size,
                              hipStream_t stream)
{
  (void)in_sizes; (void)n_in; (void)out_size; (void)ws_size;
  const float* query   = (const float*)d_in[0];
  const float* key_t   = (const float*)d_in[1];
  const float* value   = (const float*)d_in[2];
  const float* context = (const float*)d_in[3];
  const float* Wq = (const float*)d_in[4];
  const float* Wk = (const float*)d_in[5];
  const float* Wv = (const float*)d_in[6];
  const float* Wo = (const float*)d_in[7];
  const float* bo = (const float*)d_in[8];
  const float* Wc = (const float*)d_in[9];
  const float* bc = (const float*)d_in[10];
  const float* Wg = (const float*)d_in[11];
  const float* bg = (const float*)d_in[12];
  const float* opt_bias = (const float*)d_in[13];

  float* ws = (float*)d_ws;
  const long long MD = 4096LL * 1024;   // B*S*D
  const long long SS = 1048576LL;       // S*S == S*D
  float* Qp      = ws;
  float* Kp      = ws + MD;
  float* Vp      = ws + 2 * MD;
  float* gate    = ws + 3 * MD;
  float* ctx_out = gate;                // gate dead after greduce -> reuse
  float* cproj   = ws + 4 * MD;
  float* gctx    = cproj + 4096;
  float* gbuf    = gctx + 4096;
  float* rowb    = gbuf + 65536;

  float* outp = (float*)d_out;
  float* attn = outp + MD;              // second output: [B,H,S,S]

  const Epi  e0{nullptr, nullptr, 1024, 0, nullptr, 1.0f};
  const ZMap z0{0, 0, 0, 0, 0, 0, 0, 0, 0};
  dim3 blk(256, 1, 1);

  // Q/K/V projections: [4096,1024] @ [1024,1024], block tile 128x128
  {
    dim3 g(1024 / 128, 4096 / 128, 1);
    gemm_wmma_f32<4, 2, false, 0><<<g, blk, 0, stream>>>(query, 1024, Wq, 1024, Qp, 1024, 1024, e0, z0);
    gemm_wmma_f32<4, 2, false, 0><<<g, blk, 0, stream>>>(key_t, 1024, Wk, 1024, Kp, 1024, 1024, e0, z0);
    gemm_wmma_f32<4, 2, false, 0><<<g, blk, 0, stream>>>(value, 1024, Wv, 1024, Vp, 1024, 1024, e0, z0);
  }
  ctxproj_kernel<<<dim3(16), blk, 0, stream>>>(context, Wc, bc, cproj);
  gctx_kernel<<<dim3(16), blk, 0, stream>>>(cproj, Wg, gctx);
  // gate = sigmoid(query @ Wg_top + gctx[b] + bg)
  {
    const Epi eg{bg, gctx, 1024, 1024, nullptr, 1.0f};
    dim3 g(1024 / 128, 4096 / 128, 1);
    gemm_wmma_f32<4, 2, false, 2><<<g, blk, 0, stream>>>(query, 1024, Wg, 1024, gate, 1024, 1024, eg, z0);
  }
  greduce_kernel<<<dim3(256), blk, 0, stream>>>(gate, gbuf);
  rowbias_kernel<<<dim3(256), blk, 0, stream>>>(Qp, cproj, gbuf, opt_bias, rowb);
  // scores = Qh @ Kh^T / 8 + rowbias  -> written directly into attn output region
  {
    const Epi  es{nullptr, nullptr, 1024, 0, rowb, 0.125f};
    const ZMap zs{16, SS, 64, SS, 64, 16 * SS, SS, 16384LL, 1024LL};
    dim3 g(1024 / 128, 1024 / 128, 64);
    gemm_wmma_f32<4, 2, true, 3><<<g, blk, 0, stream>>>(Qp, 1024, Kp, 1024, attn, 1024, 64, es, zs);
  }
  softmax_kernel<<<dim3(65536), blk, 0, stream>>>(attn);
  // ctx_out[b,:,h*64:] = attn[b,h] @ Vh   (M=1024, N=64, K=1024), block tile 256x64
  {
    const ZMap zv{16, 16 * SS, SS, SS, 64, SS, 64, 0, 0};
    dim3 g(1, 1024 / 256, 64);
    gemm_wmma_f32<8, 1, false, 0><<<g, blk, 0, stream>>>(attn, 1024, Vp, 1024, ctx_out, 1024, 1024, e0, zv);
  }
  // out = ctx_out @ Wo + bo
  {
    const Epi eo{bo, nullptr, 1024, 0, nullptr, 1.0f};
    dim3 g(1024 / 128, 4096 / 128, 1);
    gemm_wmma_f32<4, 2, false, 1><<<g, blk, 0, stream>>>(ctx_out, 1024, Wo, 1024, outp, 1024, 1024, eo, z0);
  }
}